// QNQGRU_23794118820020
// MI455X (gfx1250) — compile-verified
//
#include <hip/hip_runtime.h>

// GRU: T=2048, B=64, I=H=256, L=2.  fp32 throughout, fp32 WMMA 16x16x4.
// Recurrent kernel uses the Tensor Data Mover (TDM) to stage h and to
// prefetch gi gate tiles into LDS (TENSORcnt / s_wait_tensorcnt path).

#define TSEQ 2048
#define BB   64
#define HH   256
#define G3   768            // 3*H
#define KD   256
#define MROWS (TSEQ * BB)   // 131072
#define NW   16             // recurrent workgroups (16 hidden cols each)

typedef __attribute__((ext_vector_type(2))) float v2f;
typedef __attribute__((ext_vector_type(8))) float v8f;
typedef unsigned int u32x4 __attribute__((ext_vector_type(4)));
typedef int i32x4 __attribute__((ext_vector_type(4)));
typedef int i32x8 __attribute__((ext_vector_type(8)));

// LDS byte offset of a __shared__ object (generic -> AS(3) -> int)
__device__ __forceinline__ unsigned lds_off(const void* p)
{
    return (unsigned)(unsigned long long)(__attribute__((address_space(3))) const char*)p;
}

// Build a D# (groups 0..3) and issue TENSOR_LOAD_TO_LDS (6-arg toolchain form).
// dims/tiles/strides in element (4-byte) units, up to 3D.
// padIC/padAC per ISA encoding (interval: 7 -> 256 dwords, amount: 3 -> 4 dwords).
__device__ __forceinline__ void tdm_load(const void* gsrc, unsigned ldsaddr,
                                         unsigned dim0, unsigned dim1, unsigned dim2,
                                         unsigned tile0, unsigned tile1, unsigned tile2,
                                         unsigned stride0, unsigned stride1,
                                         unsigned padIC, unsigned padAC, bool padEn)
{
    const unsigned long long ga = (unsigned long long)gsrc;
    u32x4 g0;
    g0[0] = 1u;                                   // count=1, user descriptor
    g0[1] = ldsaddr;                              // lds_addr
    g0[2] = (unsigned)ga;                         // global_addr[31:0]
    g0[3] = (unsigned)((ga >> 32) & 0x01FFFFFFu)  // global_addr[56:32]
          | 0x80000000u;                          // type=2 ("image")
    i32x8 g1;
    g1[0] = (int)((2u << 16) |                    // data_size = 4B
                  (padEn ? ((1u << 20) | (padIC << 22) | (padAC << 25)) : 0u));
    g1[1] = (int)((dim0 & 0xFFFFu) << 16);                      // tensor_dim0 lo
    g1[2] = (int)((dim0 >> 16) | ((dim1 & 0xFFFFu) << 16));     // dim0 hi | dim1 lo
    g1[3] = (int)((dim1 >> 16) | (tile0 << 16));                // dim1 hi | tile_dim0
    g1[4] = (int)(tile1 | (tile2 << 16));                       // tile_dim1 | tile_dim2
    g1[5] = (int)stride0;                                       // dim0_stride lo32
    g1[6] = (int)((stride1 & 0xFFFFu) << 16);                   // dim1_stride lo16
    g1[7] = (int)(stride1 >> 16);                               // dim1_stride hi
    i32x4 g2;
    g2[0] = (int)dim2;                                          // tensor_dim2
    g2[1] = 0; g2[2] = 0; g2[3] = 0;
    i32x4 g3; g3[0] = 0; g3[1] = 0; g3[2] = 0; g3[3] = 0;
    i32x8 g4; g4[0] = 0; g4[1] = 0; g4[2] = 0; g4[3] = 0;
              g4[4] = 0; g4[5] = 0; g4[6] = 0; g4[7] = 0;
    __builtin_amdgcn_tensor_load_to_lds(g0, g1, g2, g3, g4, 0);
}

// ---------------------------------------------------------------------------
// Phase GEMM: out[M,768] = A[M,256] @ W[768,256]^T + bias  (fp32 WMMA)
// ---------------------------------------------------------------------------
__global__ void __launch_bounds__(256)
gru_gate_gemm(const float* __restrict__ A, const float* __restrict__ W,
              const float* __restrict__ bias, float* __restrict__ out)
{
    __shared__ float wlds[32][KD + 4];
    __shared__ float blds[32];

    const int tid  = threadIdx.x;
    const int col0 = blockIdx.x * 32;
    const long long row0 = (long long)blockIdx.y * 128;

    for (int i = tid; i < 32 * 64; i += 256) {
        const int r  = i >> 6;
        const int c4 = i & 63;
        const float4 v = ((const float4*)(W + (long long)(col0 + r) * KD))[c4];
        *(float4*)&wlds[r][c4 * 4] = v;
    }
    if (tid < 32) blds[tid] = bias[col0 + tid];
    __syncthreads();

    const int lane = tid & 31;
    const int wv   = tid >> 5;
    const int lrow = lane & 15;
    const int kpar = (lane >> 4) << 1;
    const long long arow = row0 + wv * 16 + lrow;
    const float* aptr = A + arow * (long long)KD;

    v8f acc0 = {};
    v8f acc1 = {};
    #pragma unroll 4
    for (int k0 = 0; k0 < KD; k0 += 4) {
        const v2f a  = *(const v2f*)(aptr + k0 + kpar);
        const v2f b0 = *(const v2f*)&wlds[lrow][k0 + kpar];
        const v2f b1 = *(const v2f*)&wlds[16 + lrow][k0 + kpar];
        acc0 = __builtin_amdgcn_wmma_f32_16x16x4_f32(false, a, false, b0,
                                                     (short)0, acc0, false, false);
        acc1 = __builtin_amdgcn_wmma_f32_16x16x4_f32(false, a, false, b1,
                                                     (short)0, acc1, false, false);
    }

    const int rbump = (lane >> 4) << 3;
    float* obase = out + (row0 + wv * 16 + rbump) * (long long)G3 + col0 + lrow;
    #pragma unroll
    for (int v = 0; v < 8; ++v) {
        obase[(long long)v * G3]      = acc0[v] + blds[lrow];
        obase[(long long)v * G3 + 16] = acc1[v] + blds[16 + lrow];
    }
}

// ---------------------------------------------------------------------------
// Persistent recurrent kernel: 16 WGs, WG k owns hidden cols [16k, 16k+16).
// TDM stages h[64,256] (with LDS padding to 260-word rows) and prefetches the
// gi gate tile for step t+1 while step t computes 12 waves x 64 fp32 WMMA.
// ---------------------------------------------------------------------------
__global__ void __launch_bounds__(384)
gru_recurrent(const float* __restrict__ gi,   // [T,B,768] precomputed + b_ih
              const float* __restrict__ whh,  // [768,256]
              const float* __restrict__ bhh,  // [768]
              const float* __restrict__ h0,   // [B,256]
              float* __restrict__ hbuf,       // [2,B,256] global exchange
              float* __restrict__ y,          // [T,B,256]
              float* __restrict__ hfin,       // [B,256]
              unsigned int* __restrict__ ctr)
{
    __shared__ float hlds[BB][HH + 4];      // 64 x 260 (TDM pad: 256dw + 4dw)
    __shared__ float wlds[48][HH + 4];      // weights, resident all steps
    __shared__ float glds[3][BB][16];       // gh exchange between gate waves
    __shared__ float gilds[2][BB][3][16];   // gi tiles, TDM double buffer
    __shared__ float blds[48];

    const int tid = threadIdx.x;
    const int wg  = blockIdx.x;
    const int j0  = wg * 16;

    const unsigned hlds_a   = lds_off(&hlds[0][0]);
    const unsigned gilds_a0 = lds_off(&gilds[0][0][0][0]);
    const unsigned gilds_a1 = lds_off(&gilds[1][0][0][0]);

    // weights rows {g*256 + j0 + n} -> wlds[g*16+n][:]
    for (int i = tid; i < 48 * 64; i += 384) {
        const int lr = i >> 6;
        const int c4 = i & 63;
        const int g = lr >> 4, n = lr & 15;
        const float4 v =
            ((const float4*)(whh + (long long)(g * HH + j0 + n) * KD))[c4];
        *(float4*)&wlds[lr][c4 * 4] = v;
    }
    if (tid < 48) {
        const int g = tid >> 4, n = tid & 15;
        blds[tid] = bhh[g * HH + j0 + n];
    }

    const int lane  = tid & 31;
    const int wv    = tid >> 5;             // 0..11
    const int mt    = wv & 3;               // M tile (16 batch rows)
    const int gt    = wv >> 2;              // gate 0=r,1=z,2=n
    const int lrow  = lane & 15;
    const int kpar  = (lane >> 4) << 1;
    const int arow  = mt * 16 + lrow;
    const int wrow  = gt * 16 + lrow;
    const int rbump = (lane >> 4) << 3;

    // prefetch gi tile for t=0: 3D gather x=16 cols, y=3 gates (s=256),
    // z=64 batch rows (s=768)
    if (wv == 0) {
        tdm_load(gi + j0, gilds_a0, 16, 3, 64, 16, 3, 64, 256, 768, 0, 0, false);
    }

    // init h double-buffer slot 0 with this WG's slice of h0
    for (int i = tid; i < BB * 16; i += 384) {
        const int r = i >> 4, c = i & 15;
        hbuf[r * HH + j0 + c] = h0[r * HH + j0 + c];
    }
    __threadfence();
    __syncthreads();
    if (tid == 0) {
        __hip_atomic_fetch_add(ctr, 1u, __ATOMIC_RELEASE, __HIP_MEMORY_SCOPE_AGENT);
        while (__hip_atomic_load(ctr, __ATOMIC_ACQUIRE, __HIP_MEMORY_SCOPE_AGENT) < NW)
            __builtin_amdgcn_s_sleep(1);
    }
    __syncthreads();

    for (int t = 0; t < TSEQ; ++t) {
        __threadfence();                    // acquire other WGs' h writes
        const float* hsrc = hbuf + (t & 1) * (BB * HH);
        float*       hdst = hbuf + ((t + 1) & 1) * (BB * HH);

        if (wv == 0) {
            // TDM: h[64,256] -> hlds with 4-dword row padding
            tdm_load(hsrc, hlds_a, HH, BB, 0, HH, BB, 0, HH, 0, 7u, 3u, true);
            if (t + 1 < TSEQ) {
                // TDM: prefetch gi tile for next step
                tdm_load(gi + (long long)(t + 1) * (BB * G3) + j0,
                         ((t + 1) & 1) ? gilds_a1 : gilds_a0,
                         16, 3, 64, 16, 3, 64, 256, 768, 0, 0, false);
                __builtin_amdgcn_s_wait_tensorcnt(1);  // gi(t), h(t) complete
            } else {
                __builtin_amdgcn_s_wait_tensorcnt(0);
            }
        }
        __syncthreads();

        // gh tile: [16 rows] x [16 cols of gate gt], K = 256
        v8f acc = {};
        #pragma unroll 8
        for (int k0 = 0; k0 < HH; k0 += 4) {
            const v2f a = *(const v2f*)&hlds[arow][k0 + kpar];
            const v2f b = *(const v2f*)&wlds[wrow][k0 + kpar];
            acc = __builtin_amdgcn_wmma_f32_16x16x4_f32(false, a, false, b,
                                                        (short)0, acc, false, false);
        }
        #pragma unroll
        for (int v = 0; v < 8; ++v)
            glds[gt][mt * 16 + rbump + v][lrow] = acc[v] + blds[gt * 16 + lrow];
        __syncthreads();

        // combine gates -> h_new[:, j0:j0+16]  (pure LDS + VALU)
        const float (*git)[3][16] = gilds[t & 1];
        for (int i = tid; i < BB * 16; i += 384) {
            const int r = i >> 4, c = i & 15;
            const float rg = 1.f / (1.f + __expf(-(git[r][0][c] + glds[0][r][c])));
            const float zg = 1.f / (1.f + __expf(-(git[r][1][c] + glds[1][r][c])));
            const float ng = tanhf(git[r][2][c] + rg * glds[2][r][c]);
            const float hprev = hlds[r][j0 + c];
            const float hy = ng + zg * (hprev - ng);
            hdst[r * HH + j0 + c] = hy;
            y[(long long)t * (BB * HH) + r * HH + j0 + c] = hy;
            if (t == TSEQ - 1) hfin[r * HH + j0 + c] = hy;
        }
        __threadfence();                    // release h_new before arriving
        __syncthreads();
        if (tid == 0) {
            __hip_atomic_fetch_add(ctr, 1u, __ATOMIC_RELEASE, __HIP_MEMORY_SCOPE_AGENT);
            const unsigned tgt = (unsigned)(t + 2) * NW;
            while (__hip_atomic_load(ctr, __ATOMIC_ACQUIRE, __HIP_MEMORY_SCOPE_AGENT) < tgt)
                __builtin_amdgcn_s_sleep(1);
        }
        __syncthreads();
    }
}

// ---------------------------------------------------------------------------
extern "C" void kernel_launch(void* const* d_in, const int* in_sizes, int n_in,
                              void* d_out, int out_size, void* d_ws, size_t ws_size,
                              hipStream_t stream)
{
    const float* x     = (const float*)d_in[0];
    const float* h0    = (const float*)d_in[1];   // [2,B,H]
    const float* w_ih0 = (const float*)d_in[2];
    const float* w_hh0 = (const float*)d_in[3];
    const float* b_ih0 = (const float*)d_in[4];
    const float* b_hh0 = (const float*)d_in[5];
    const float* w_ih1 = (const float*)d_in[6];
    const float* w_hh1 = (const float*)d_in[7];
    const float* b_ih1 = (const float*)d_in[8];
    const float* b_hh1 = (const float*)d_in[9];
    (void)in_sizes; (void)n_in; (void)out_size; (void)ws_size;

    char* ws = (char*)d_ws;
    unsigned int* ctr = (unsigned int*)ws;                       // 2 counters
    float* gi   = (float*)(ws + 256);                            // [T*B,768] (reused)
    float* y0   = gi + (size_t)MROWS * G3;                       // [T*B,256]
    float* hbuf = y0 + (size_t)MROWS * HH;                       // [2,B,256]

    float* out = (float*)d_out;
    float* y1  = out;                                            // [T,B,H]
    float* hf0 = out + (size_t)MROWS * HH;                       // h_final[0]
    float* hf1 = hf0 + BB * HH;                                  // h_final[1]

    (void)hipMemsetAsync(ctr, 0, 256, stream);

    const dim3 gg(G3 / 32, MROWS / 128);
    // layer 0
    gru_gate_gemm<<<gg, 256, 0, stream>>>(x, w_ih0, b_ih0, gi);
    gru_recurrent<<<NW, 384, 0, stream>>>(gi, w_hh0, b_hh0, h0, hbuf, y0, hf0, ctr);
    // layer 1 (gi buffer reused)
    gru_gate_gemm<<<gg, 256, 0, stream>>>(y0, w_ih1, b_ih1, gi);
    gru_recurrent<<<NW, 384, 0, stream>>>(gi, w_hh1, b_hh1, h0 + BB * HH,
                                          hbuf, y1, hf1, ctr + 1);
}